// TutteLayer_9371618640224
// MI455X (gfx1250) — compile-verified
//
#include <hip/hip_runtime.h>
#include <math.h>

#define GRID 64
#define NV   4096
#define NB   252
#define NI   3844          // 62*62 interior vertices
#define MTILES 241         // ceil(3844/16)

// -------- workspace layout (float offsets) --------
#define WS_WE   0          // 4096  edge weight to (i, j+1)
#define WS_WSS  4096       // 4096  edge weight to (i+1, j)
#define WS_WSE  8192       // 4096  edge weight to (i+1, j+1)
#define WS_D    12288      // 4096  diagonal (full degree)
#define WS_BP   16384      // 512   boundary positions (252*2 used)
#define WS_B    16896      // 7712  rhs b (3844*2 used)
#define WS_BV   24608      // 256   boundary ordinal -> vertex id (int), -1000 pad

// -------- output layout (float offsets): pred | out_pos | dist --------
#define OFF_OUTPOS 524288
#define OFF_DIST   532480

typedef __attribute__((ext_vector_type(2))) float v2f;
typedef __attribute__((ext_vector_type(8))) float v8f;
typedef __attribute__((ext_vector_type(4))) unsigned int u32x4;
typedef __attribute__((ext_vector_type(4))) int i32x4;
typedef __attribute__((ext_vector_type(8))) int i32x8;

__global__ void k_zero(float* ws, int n) {
    int t = blockIdx.x * blockDim.x + threadIdx.x;
    if (t < n) ws[t] = 0.f;
}

// Scatter squashed edge weights into the three stencil planes (one writer per slot).
__global__ void k_edges(const float* __restrict__ Wvar, const int* __restrict__ edges,
                        float* __restrict__ ws, int E) {
    int e = blockIdx.x * blockDim.x + threadIdx.x;
    if (e >= E) return;
    int u = edges[2 * e], v = edges[2 * e + 1];      // u < v (unique sorted pairs)
    float w = 0.6f / (1.f + expf(-Wvar[e])) + 0.2f;  // sigmoid*0.6+0.2
    int dv = v - u;
    if (dv == 1)          ws[WS_WE  + u] = w;
    else if (dv == GRID)  ws[WS_WSS + u] = w;
    else                  ws[WS_WSE + u] = w;        // dv == GRID+1
}

// Deterministic diagonal: sum of the 6 incident weights (missing edges are 0).
__global__ void k_degree(float* __restrict__ ws) {
    int v = blockIdx.x * blockDim.x + threadIdx.x;
    if (v >= NV) return;
    int gi = v >> 6, gj = v & 63;
    float d = ws[WS_WE + v] + ws[WS_WSS + v] + ws[WS_WSE + v];
    if (gj > 0)           d += ws[WS_WE  + v - 1];
    if (gi > 0)           d += ws[WS_WSS + v - 64];
    if (gi > 0 && gj > 0) d += ws[WS_WSE + v - 65];
    ws[WS_D + v] = d;
}

// Boundary: sigmoid, normalized inclusive scan, angle->unit-square map.
// Also publishes the ordinal->vertex table used by the WMMA rhs kernel.
__global__ void k_boundary(const float* __restrict__ avar, const int* __restrict__ bverts,
                           float* __restrict__ ws, float* __restrict__ out) {
    __shared__ float s[256];
    int t = threadIdx.x;
    int* wsI = (int*)ws;
    wsI[WS_BV + t] = (t < NB) ? bverts[t] : -1000;   // sentinel never matches a nbr
    s[t] = (t < NB) ? (0.6f / (1.f + expf(-avar[t])) + 0.2f) : 0.f;
    __syncthreads();
    for (int off = 1; off < 256; off <<= 1) {       // Hillis-Steele inclusive scan
        float w = (t >= off) ? s[t - off] : 0.f;
        __syncthreads();
        s[t] += w;
        __syncthreads();
    }
    if (t < NB) {
        const float PI = 3.14159265358979323846f;
        float angle = s[t] / s[NB - 1] * (2.f * PI);
        float tn = tanf(angle);
        float bx, by;
        if ((angle > 7.f * PI / 4.f) || (angle <= PI / 4.f))      { bx = 1.f;       by = tn;   }
        else if ((angle > PI / 4.f) && (angle <= 3.f * PI / 4.f)) { bx = 1.f / tn;  by = 1.f;  }
        else if ((angle > 3.f*PI/4.f) && (angle <= 5.f*PI/4.f))   { bx = -1.f;      by = -tn;  }
        else                                                      { bx = -1.f / tn; by = -1.f; }
        ws[WS_BP + 2 * t] = bx; ws[WS_BP + 2 * t + 1] = by;
        int bv = bverts[t];
        out[OFF_OUTPOS + 2 * bv]     = bx;           // fixed boundary positions
        out[OFF_OUTPOS + 2 * bv + 1] = by;
    }
}

// Branchless A[m][k]: one table lookup for the boundary vertex, six disjoint
// arithmetic 0/1 flags, address as a sum of products, one load, one select.
__device__ __forceinline__ float a_val(const float* __restrict__ ws,
                                       const int* __restrict__ wsI,
                                       int mok, int gi, int gj, int v, int k) {
    int bv = wsI[WS_BV + k];                 // boundary vertex id (or -1000 pad)
    int di = (bv >> 6) - gi;
    int dj = (bv & 63) - gj;
    int eE  = (int)(di == 0)  * (int)(dj == 1);
    int eW  = (int)(di == 0)  * (int)(dj == -1);
    int eS  = (int)(di == 1)  * (int)(dj == 0);
    int eN  = (int)(di == -1) * (int)(dj == 0);
    int eSE = (int)(di == 1)  * (int)(dj == 1);
    int eNW = (int)(di == -1) * (int)(dj == -1);
    int hit = (eE + eW + eS + eN + eSE + eNW) * mok;
    int adx = eE  * (WS_WE  + v)      + eW  * (WS_WE  + v - 1)
            + eS  * (WS_WSS + v)      + eN  * (WS_WSS + v - 64)
            + eSE * (WS_WSE + v)      + eNW * (WS_WSE + v - 65);
    float w = ws[adx];                       // adx==0 when no flag set: valid slot
    return hit ? w : 0.f;
}

__device__ __forceinline__ float b_val(const float* __restrict__ ws, int k, int n) {
    bool hit = (k < NB) && (n < 2);
    float w = ws[hit ? (WS_BP + 2 * k + n) : 0];    // masked address, no branch
    return hit ? w : 0.f;
}

// rhs b = A_ib * bound_pos via V_WMMA_F32_16X16X4_F32.  One wave per 16-row tile,
// K = 252 padded to 256 -> 64 k-steps, unrolled x8.  EXEC stays all-ones.
__global__ void k_rhs(float* __restrict__ ws) {
    const int* wsI  = (const int*)ws;
    const int tile  = blockIdx.x;
    const int lane  = threadIdx.x;        // 32-thread block
    const int n     = lane & 15;          // column / A-row within tile
    const int khalf = lane >> 4;          // lanes 16-31 hold K+2,K+3
    const int m     = tile * 16 + n;
    const int mok   = (int)(m < NI);
    const int mm    = mok ? m : 0;
    const int gi    = mm / 62 + 1;
    const int gj    = mm % 62 + 1;
    const int v     = gi * 64 + gj;
    v8f c = {0.f, 0.f, 0.f, 0.f, 0.f, 0.f, 0.f, 0.f};
    #pragma unroll 8
    for (int s = 0; s < 64; ++s) {
        int klo = s * 4 + khalf * 2;
        v2f a, b;
        a.x = a_val(ws, wsI, mok, gi, gj, v, klo);
        a.y = a_val(ws, wsI, mok, gi, gj, v, klo + 1);
        b.x = b_val(ws, klo, n);
        b.y = b_val(ws, klo + 1, n);
        c = __builtin_amdgcn_wmma_f32_16x16x4_f32(false, a, false, b, (short)0, c,
                                                  false, false);
    }
    if (n < 2) {                          // only columns 0,1 are meaningful
        int mb = khalf * 8;
        #pragma unroll
        for (int r = 0; r < 8; ++r) {
            int row = tile * 16 + mb + r;
            if (row < NI) ws[WS_B + 2 * row + n] = c[r];
        }
    }
}

__device__ __forceinline__ float block_reduce(float v, float* sred) {
    #pragma unroll
    for (int o = 16; o > 0; o >>= 1) v += __shfl_down(v, o, 32);
    int wid = threadIdx.x >> 5, lane = threadIdx.x & 31;
    if (lane == 0) sred[wid] = v;
    __syncthreads();
    if (wid == 0) {
        float s2 = sred[lane];
        #pragma unroll
        for (int o = 16; o > 0; o >>= 1) s2 += __shfl_down(s2, o, 32);
        if (lane == 0) sred[0] = s2;
    }
    __syncthreads();
    float r = sred[0];
    __syncthreads();
    return r;
}

// Single-WGP LDS-resident CG on the 62x62 stencil.  Coefficients arrive via TDM.
__global__ void __launch_bounds__(1024, 1)
k_cg(float* __restrict__ ws, float* __restrict__ out, int iters) {
    extern __shared__ float lds[];
    float2* sp   = (float2*)lds;          // p-vector, 4096 float2   [0 .. 8192)
    float*  sW   = lds + 8192;            // WE|WS|WSE|D planes      [8192 .. 24576)
    float*  sred = lds + 24576;           // reduction scratch (64)
    const int t = threadIdx.x;

    // --- Tensor Data Mover: ws[0..16384) floats -> LDS byte 32768, 64 KB, 1 row ---
    if (t < 32) {                          // one wave issues the DMA
        unsigned long long ga = (unsigned long long)(uintptr_t)ws;
        u32x4 g0;
        g0.x = 1u;                                             // count=1 (valid D#)
        g0.y = 8192u * 4u;                                     // lds_addr (bytes)
        g0.z = (unsigned)(ga & 0xFFFFFFFFu);                   // global_addr lo
        g0.w = (unsigned)((ga >> 32) & 0x01FFFFFFu) | (2u << 30); // addr hi | type=2
        i32x8 g1;
        g1[0] = (int)(2u << 16);          // data_size = 4B
        g1[1] = (int)(16384u << 16);      // tensor_dim0 = 16384 (low 16 in [31:16])
        g1[2] = (int)(1u << 16);          // tensor_dim1 = 1
        g1[3] = (int)(16384u << 16);      // tile_dim0 = 16384
        g1[4] = 1;                        // tile_dim1 = 1
        g1[5] = 16384;                    // tensor_dim0_stride
        g1[6] = 0; g1[7] = 0;
        i32x4 z4 = {0, 0, 0, 0};
#if defined(__clang_major__) && __clang_major__ >= 23
        i32x8 z8 = {0, 0, 0, 0, 0, 0, 0, 0};
        __builtin_amdgcn_tensor_load_to_lds(g0, g1, z4, z4, z8, 0);
#else
        __builtin_amdgcn_tensor_load_to_lds(g0, g1, z4, z4, 0);
#endif
    }

    // --- init CG state while the TDM transfer is in flight ---
    float2 x[4], r[4];
    float rr_loc = 0.f;
    #pragma unroll
    for (int k2 = 0; k2 < 4; ++k2) {
        int c = t + k2 * 1024;
        int gi = c >> 6, gj = c & 63;
        bool inter = (gi >= 1) && (gi <= 62) && (gj >= 1) && (gj <= 62);
        float2 bv = {0.f, 0.f};
        if (inter) {
            int ridx = (gi - 1) * 62 + (gj - 1);
            bv.x = ws[WS_B + 2 * ridx];
            bv.y = ws[WS_B + 2 * ridx + 1];
        }
        x[k2].x = 0.f; x[k2].y = 0.f;
        r[k2] = bv;
        sp[c] = bv;                        // p = r = b ; boundary cells stay 0
        rr_loc += bv.x * bv.x + bv.y * bv.y;
    }
    if (t < 32) __builtin_amdgcn_s_wait_tensorcnt(0);
    __syncthreads();
    float rr = block_reduce(rr_loc, sred);

    for (int it = 0; it < iters; ++it) {
        float2 ap[4], pc[4];
        float pap_loc = 0.f;
        #pragma unroll
        for (int k2 = 0; k2 < 4; ++k2) {
            int c = t + k2 * 1024;
            int gi = c >> 6, gj = c & 63;
            bool inter = (gi >= 1) && (gi <= 62) && (gj >= 1) && (gj <= 62);
            float2 pcc = sp[c];  pc[k2] = pcc;
            float2 a = {0.f, 0.f};
            if (inter) {
                float wE  = sW[c],          wEm = sW[c - 1];
                float wS  = sW[4096 + c],   wSm = sW[4096 + c - 64];
                float wD  = sW[8192 + c],   wDm = sW[8192 + c - 65];
                float dg  = sW[12288 + c];
                float2 pE = sp[c + 1],  pW  = sp[c - 1];
                float2 pS = sp[c + 64], pN  = sp[c - 64];
                float2 pD = sp[c + 65], pDm = sp[c - 65];
                a.x = dg*pcc.x - wE*pE.x - wEm*pW.x - wS*pS.x - wSm*pN.x - wD*pD.x - wDm*pDm.x;
                a.y = dg*pcc.y - wE*pE.y - wEm*pW.y - wS*pS.y - wSm*pN.y - wD*pD.y - wDm*pDm.y;
            }
            ap[k2] = a;
            pap_loc += pcc.x * a.x + pcc.y * a.y;
        }
        float pap   = block_reduce(pap_loc, sred);
        float alpha = (pap != 0.f) ? rr / pap : 0.f;
        float rrn_loc = 0.f;
        #pragma unroll
        for (int k2 = 0; k2 < 4; ++k2) {
            x[k2].x += alpha * pc[k2].x;  x[k2].y += alpha * pc[k2].y;
            r[k2].x -= alpha * ap[k2].x;  r[k2].y -= alpha * ap[k2].y;
            rrn_loc += r[k2].x * r[k2].x + r[k2].y * r[k2].y;
        }
        float rrn  = block_reduce(rrn_loc, sred);   // barrier => all p reads done
        float beta = (rr != 0.f) ? rrn / rr : 0.f;
        rr = rrn;
        #pragma unroll
        for (int k2 = 0; k2 < 4; ++k2) {
            int c = t + k2 * 1024;
            float2 np;
            np.x = r[k2].x + beta * pc[k2].x;
            np.y = r[k2].y + beta * pc[k2].y;
            sp[c] = np;
        }
        __syncthreads();
    }
    #pragma unroll
    for (int k2 = 0; k2 < 4; ++k2) {
        int c = t + k2 * 1024;
        int gi = c >> 6, gj = c & 63;
        if (gi >= 1 && gi <= 62 && gj >= 1 && gj <= 62) {
            out[OFF_OUTPOS + 2 * c]     = x[k2].x;
            out[OFF_OUTPOS + 2 * c + 1] = x[k2].y;
        }
    }
}

// Per-point: barycentric blend + closed-form 2x2 affine distortion.
__global__ void k_points(const float* __restrict__ pts, const int* __restrict__ faces,
                         const int* __restrict__ fids, const float* __restrict__ verts,
                         const float* __restrict__ opos, float* __restrict__ out, int N) {
    int p = blockIdx.x * blockDim.x + threadIdx.x;
    if (p >= N) return;
    float lx = pts[2 * p], ly = pts[2 * p + 1];
    int f  = fids[p];
    int t0 = faces[3 * f], t1 = faces[3 * f + 1], t2 = faces[3 * f + 2];
    float o0x = verts[2 * t0], o0y = verts[2 * t0 + 1];
    float o1x = verts[2 * t1], o1y = verts[2 * t1 + 1];
    float o2x = verts[2 * t2], o2y = verts[2 * t2 + 1];
    float n0x = opos[2 * t0],  n0y = opos[2 * t0 + 1];
    float n1x = opos[2 * t1],  n1y = opos[2 * t1 + 1];
    float n2x = opos[2 * t2],  n2y = opos[2 * t2 + 1];
    float aA = fabsf((lx - o1x) * (ly - o2y) - (ly - o1y) * (lx - o2x)) * 0.5f;
    float aB = fabsf((lx - o0x) * (ly - o2y) - (ly - o0y) * (lx - o2x)) * 0.5f;
    float aC = fabsf((lx - o0x) * (ly - o1y) - (ly - o0y) * (lx - o1x)) * 0.5f;
    float inv = 1.f / (aA + aB + aC);
    out[2 * p]     = (n0x * aA + n1x * aB + n2x * aC) * inv;
    out[2 * p + 1] = (n0y * aA + n1y * aB + n2y * aC) * inv;
    float e1ox = o1x - o0x, e1oy = o1y - o0y, e2ox = o2x - o0x, e2oy = o2y - o0y;
    float e1nx = n1x - n0x, e1ny = n1y - n0y, e2nx = n2x - n0x, e2ny = n2y - n0y;
    float id = 1.f / (e1ox * e2oy - e2ox * e1oy);
    out[OFF_DIST + 4 * p + 0] = (e1nx * e2oy - e2nx * e1oy) * id;
    out[OFF_DIST + 4 * p + 1] = (e2nx * e1ox - e1nx * e2ox) * id;
    out[OFF_DIST + 4 * p + 2] = (e1ny * e2oy - e2ny * e1oy) * id;
    out[OFF_DIST + 4 * p + 3] = (e2ny * e1ox - e1ny * e2ox) * id;
}

extern "C" void kernel_launch(void* const* d_in, const int* in_sizes, int n_in,
                              void* d_out, int out_size, void* d_ws, size_t ws_size,
                              hipStream_t stream) {
    const float* pts   = (const float*)d_in[0];
    const float* Wvar  = (const float*)d_in[1];
    const float* avar  = (const float*)d_in[2];
    const float* verts = (const float*)d_in[3];
    const int*   faces = (const int*)d_in[4];
    const int*   edges = (const int*)d_in[5];
    const int*   fids  = (const int*)d_in[6];
    const int*   bvrt  = (const int*)d_in[7];
    float* out = (float*)d_out;
    float* ws  = (float*)d_ws;
    int E = in_sizes[5] / 2;
    int N = in_sizes[0] / 2;

    k_zero    <<<(16384 + 255) / 256, 256, 0, stream>>>(ws, 16384);
    k_edges   <<<(E + 255) / 256,     256, 0, stream>>>(Wvar, edges, ws, E);
    k_degree  <<<(NV + 255) / 256,    256, 0, stream>>>(ws);
    k_boundary<<<1,                   256, 0, stream>>>(avar, bvrt, ws, out);
    k_rhs     <<<MTILES,               32, 0, stream>>>(ws);
    k_cg      <<<1, 1024, 24640 * sizeof(float), stream>>>(ws, out, 512);
    k_points  <<<(N + 255) / 256,     256, 0, stream>>>(pts, faces, fids, verts,
                                                        out + OFF_OUTPOS, out, N);
}